// GCNLayer_12197707120939
// MI455X (gfx1250) — compile-verified
//
#include <hip/hip_runtime.h>

typedef __attribute__((ext_vector_type(2))) float v2f;
typedef __attribute__((ext_vector_type(8))) float v8f;

#define N_NODES 100000
#define F 128

// ---------------------------------------------------------------------------
// Kernel 1: xw = x @ W   (fp32 WMMA 16x16x4, W staged in LDS pre-swizzled
// into B-fragment order so each fragment is ONE contiguous ds_load_b64).
//
// Swizzled LDS index for W[k][n]:
//   dst = ( (k>>2)*256 + ((k>>1)&1)*128 + n ) * 2 + (k&1)
// so the pair (W[k0+2h][n], W[k0+2h+1][n]) needed by lane (h, r=n&15) for
// tile t (n = t*16+r) sits at one 8-byte-aligned slot.
// A-layout (16x4 f32): lanes 0-15 -> VGPR0/1 = K0/K1; lanes 16-31 -> K2/K3,
// which is a contiguous 8B load from the row-major x row.
// C-layout: VGPR i, lanes 0-15 -> M=i, lanes 16-31 -> M=i+8, N=lane&15.
// ---------------------------------------------------------------------------
__global__ __launch_bounds__(256) void gemm_xw_kernel(const float* __restrict__ x,
                                                      const float* __restrict__ W,
                                                      float* __restrict__ xw) {
    __shared__ float Wswz[F * F];  // 64 KB (WGP has 320 KB LDS)

    {   // cooperative swizzling copy: 4096 float4 reads, scattered LDS stores
        const int t = threadIdx.x;
        #pragma unroll
        for (int i = 0; i < 16; ++i) {
            const int vidx = t + i * 256;              // float4 index (4096 total)
            const float4 w4 = ((const float4*)W)[vidx];
            const int k = vidx >> 5;                   // 32 float4 per K-row
            const int n = (vidx & 31) * 4;
            const int base = (((k >> 2) << 8) + (((k >> 1) & 1) << 7) + n) * 2 + (k & 1);
            Wswz[base + 0] = w4.x;
            Wswz[base + 2] = w4.y;
            Wswz[base + 4] = w4.z;
            Wswz[base + 6] = w4.w;
        }
    }
    __syncthreads();

    const int wave = threadIdx.x >> 5;   // 0..7
    const int lane = threadIdx.x & 31;
    const int half = lane >> 4;          // 0: K pair {0,1}, 1: K pair {2,3}
    const int r    = lane & 15;

    const int rowBase = (blockIdx.x * 8 + wave) * 16;
    int loadRow = rowBase + r;
    if (loadRow >= N_NODES) loadRow = N_NODES - 1;   // keep EXEC all-1s for WMMA
    const float* __restrict__ xrow = x + (size_t)loadRow * F;

    v8f acc[8];
    #pragma unroll
    for (int t = 0; t < 8; ++t) acc[t] = (v8f){};

    #pragma unroll 2
    for (int k0 = 0; k0 < F; k0 += 4) {
        // A fragment: contiguous 8B per lane
        const v2f a = *(const v2f*)(xrow + k0 + 2 * half);
        // B fragments: one contiguous ds_load_b64 each, bank-conflict free
        const float* __restrict__ bbase =
            &Wswz[(((k0 >> 2) << 8) + (half << 7) + r) * 2];
        #pragma unroll
        for (int t = 0; t < 8; ++t) {
            const v2f bf = *(const v2f*)(bbase + t * 32);
            acc[t] = __builtin_amdgcn_wmma_f32_16x16x4_f32(
                false, a, false, bf, (short)0, acc[t], false, false);
        }
    }

    #pragma unroll
    for (int i = 0; i < 8; ++i) {
        const int row = rowBase + i + 8 * half;
        if (row < N_NODES) {
            float* __restrict__ orow = xw + (size_t)row * F;
            #pragma unroll
            for (int t = 0; t < 8; ++t)
                orow[t * 16 + r] = acc[t][i];
        }
    }
}

// ---------------------------------------------------------------------------
// Kernel 2: out[n, :] = b   (bias folded into scatter's accumulator init)
// ---------------------------------------------------------------------------
__global__ __launch_bounds__(256) void init_bias_kernel(const float* __restrict__ b,
                                                        float* __restrict__ out) {
    const int idx = blockIdx.x * blockDim.x + threadIdx.x;   // float4 index
    const int total = N_NODES * F / 4;                       // 3,200,000
    const float4 bv = ((const float4*)b)[idx & (F / 4 - 1)]; // F/4 == 32
    if (idx < total) ((float4*)out)[idx] = bv;
}

// ---------------------------------------------------------------------------
// Kernel 3: out += A @ xw  (COO scatter, one wave per edge)
// lane i gathers xw[col][4i..4i+3] (coalesced float4) and issues 4 native
// global_atomic_add_f32 into out[row]. xw and out are L2-resident (51MB each
// vs 192MB L2), so gathers + atomics never hit HBM.
// ---------------------------------------------------------------------------
__global__ __launch_bounds__(256) void edge_scatter_kernel(const int* __restrict__ row,
                                                           const int* __restrict__ col,
                                                           const float* __restrict__ vals,
                                                           const float* __restrict__ xw,
                                                           float* __restrict__ out,
                                                           int nEdges) {
    const int lane   = threadIdx.x & 31;
    const int waveId = (blockIdx.x * blockDim.x + threadIdx.x) >> 5;
    const int nWaves = (gridDim.x * blockDim.x) >> 5;

    for (int e = waveId; e < nEdges; e += nWaves) {
        const int   rr = row[e];
        const int   cc = col[e];
        const float v  = vals[e];
        const float4 xv = ((const float4*)(xw + (size_t)cc * F))[lane];
        float* __restrict__ dst = out + (size_t)rr * F + lane * 4;
        unsafeAtomicAdd(dst + 0, v * xv.x);
        unsafeAtomicAdd(dst + 1, v * xv.y);
        unsafeAtomicAdd(dst + 2, v * xv.z);
        unsafeAtomicAdd(dst + 3, v * xv.w);
    }
}

// ---------------------------------------------------------------------------
extern "C" void kernel_launch(void* const* d_in, const int* in_sizes, int n_in,
                              void* d_out, int out_size, void* d_ws, size_t ws_size,
                              hipStream_t stream) {
    const float* x  = (const float*)d_in[0];
    const int*   rw = (const int*)d_in[1];
    const int*   cl = (const int*)d_in[2];
    const float* ev = (const float*)d_in[3];
    const float* W  = (const float*)d_in[4];
    const float* b  = (const float*)d_in[5];
    float* out = (float*)d_out;
    float* xw  = (float*)d_ws;            // N_NODES * F floats = 51.2 MB scratch
    const int nEdges = in_sizes[1];

    // 1) xw = x @ W  (dense GEMM via fp32 WMMA)
    const int gemmBlocks = (N_NODES + 127) / 128;   // 8 waves x 16 rows per block
    gemm_xw_kernel<<<gemmBlocks, 256, 0, stream>>>(x, W, xw);

    // 2) out = broadcast(b)
    init_bias_kernel<<<(N_NODES * F / 4 + 255) / 256, 256, 0, stream>>>(b, out);

    // 3) out += A @ xw  (atomic scatter; final result)
    edge_scatter_kernel<<<2048, 256, 0, stream>>>(rw, cl, ev, xw, out, nEdges);
}